// NeuralDecisionTree_8770323218482
// MI455X (gfx1250) — compile-verified
//
#include <hip/hip_runtime.h>

#define BATCH   16384
#define NFEAT   512
#define NUSED   256
#define NLEAF   1024
#define NCLS    100
#define NCLSP   112   // padded class count (7 tiles of 16)
#define TDEPTH  10

typedef __attribute__((ext_vector_type(2))) float v2f;
typedef __attribute__((ext_vector_type(8))) float v8f;

// LDS strides (chosen for bank-conflict-free access, 64 banks x 4B)
#define XST 17      // xsT[k][m], k in [0,256)
#define DST 1028    // ds[m][node], m in [0,16), node in [0,1024)
#define MUT 17      // musT[k][m], k in [0,64)

// ---------------------------------------------------------------------------
// Kernel 1: recover gather indices from the one-hot mask rows.
__global__ __launch_bounds__(256) void ndt_extract_idx(const float* __restrict__ mask,
                                                       int* __restrict__ idx) {
    int j = threadIdx.x;            // 256 rows, one thread each
    const float* row = mask + j * NFEAT;
    int found = 0;
    for (int i = 0; i < NFEAT; ++i) {
        if (row[i] > 0.5f) { found = i; break; }
    }
    idx[j] = found;
}

// ---------------------------------------------------------------------------
// Kernel 2: probs = softmax(pi, axis=-1), stored padded to NCLSP cols (zeros).
__global__ __launch_bounds__(256) void ndt_softmax_pi(const float* __restrict__ pi,
                                                      float* __restrict__ probs) {
    int r = blockIdx.x * 256 + threadIdx.x;
    if (r >= NLEAF) return;
    const float* src = pi + r * NCLS;
    float mx = -1e30f;
    for (int c = 0; c < NCLS; ++c) mx = fmaxf(mx, src[c]);
    float s = 0.f;
    for (int c = 0; c < NCLS; ++c) s += __expf(src[c] - mx);
    float inv = __builtin_amdgcn_rcpf(s);           // v_rcp_f32, ~1 ulp
    float* dst = probs + r * NCLSP;
    for (int c = 0; c < NCLS; ++c) dst[c] = __expf(src[c] - mx) * inv;
    for (int c = NCLS; c < NCLSP; ++c) dst[c] = 0.f;
}

// ---------------------------------------------------------------------------
// Kernel 3: fused  gather -> GEMM1(sigmoid) -> tree-product -> GEMM2 -> out
// One block = 16 batch rows, 256 threads = 8 waves (wave32).
__global__ __launch_bounds__(256) void ndt_main(const float* __restrict__ feat,
                                                const float* __restrict__ W,
                                                const float* __restrict__ bias,
                                                const int*   __restrict__ idx,
                                                const float* __restrict__ probs,
                                                float* __restrict__ out) {
    // ds:   16 x 1028 floats (d matrix)           = 16448 floats
    // musT: 64 x 17 floats (mu chunk, transposed) =  1088 floats
    // xsT (256 x 17 = 4352 floats) is aliased on top of ds (used disjoint in time)
    __shared__ float smem[16 * DST + 64 * MUT];   // 70144 bytes
    float* ds   = smem;
    float* xsT  = smem;                 // alias: only live before first barrier pair
    float* musT = smem + 16 * DST;

    const int tid  = threadIdx.x;
    const int wave = tid >> 5;          // 0..7 (wave-uniform)
    const int lane = tid & 31;
    const int lm   = lane & 15;         // M (or N) index within a 16-wide tile
    const int lh   = lane >> 4;         // K-half selector for A/B frags, M-half for C
    const int row0 = blockIdx.x * 16;

    // ---- Phase 1: gather transposed x tile: xsT[j][m] = feat[row0+m][idx[j]]
    {
        int m  = tid >> 4;              // 0..15
        int jb = tid & 15;
        const float* frow = feat + (size_t)(row0 + m) * NFEAT;
        #pragma unroll
        for (int i = 0; i < 16; ++i) {
            int j = jb + 16 * i;
            xsT[j * XST + m] = frow[idx[j]];
        }
    }
    __syncthreads();

    // ---- Phase 2: GEMM1: logits[16 x 1024] = x[16 x 256] * W[256 x 1024]
    // Each wave owns 8 consecutive 16-wide N tiles (128 cols).
    const v8f vzero = {0.f, 0.f, 0.f, 0.f, 0.f, 0.f, 0.f, 0.f};
    v8f acc[8];
    #pragma unroll
    for (int t = 0; t < 8; ++t) acc[t] = vzero;

    const int n0w = wave * 128;
    float bv[8];
    #pragma unroll
    for (int t = 0; t < 8; ++t) bv[t] = bias[n0w + t * 16 + lm];

    for (int k0 = 0; k0 < NUSED; k0 += 4) {
        const int ka = k0 + 2 * lh;                 // this lane's K base
        v2f a;
        a.x = xsT[ka * XST + lm];                   // A[m][ka]
        a.y = xsT[(ka + 1) * XST + lm];             // A[m][ka+1]
        const float* wp = W + (size_t)ka * NLEAF + lm;
        #pragma unroll
        for (int t = 0; t < 8; ++t) {
            const int nc = n0w + t * 16;
            v2f bf;
            bf.x = wp[nc];                          // B[ka][n]
            bf.y = wp[NLEAF + nc];                  // B[ka+1][n]
            acc[t] = __builtin_amdgcn_wmma_f32_16x16x4_f32(
                false, a, false, bf, (short)0, acc[t], false, false);
        }
    }

    // sigmoid in registers (xsT region still live for other waves)
    // 1/(1+e) via v_rcp_f32: ~1 ulp, avoids the IEEE div_scale/div_fmas sequence
    #pragma unroll
    for (int t = 0; t < 8; ++t) {
        #pragma unroll
        for (int r = 0; r < 8; ++r) {
            float z = acc[t][r] + bv[t];
            float e = __expf(-z);
            acc[t][r] = __builtin_amdgcn_rcpf(1.0f + e);
        }
    }
    __syncthreads();    // everyone done reading xsT; safe to overwrite with ds

    // ---- Phase 3: write d to LDS: ds[m][node]
    #pragma unroll
    for (int t = 0; t < 8; ++t) {
        const int nc = n0w + t * 16 + lm;
        #pragma unroll
        for (int r = 0; r < 8; ++r) {
            ds[(r + 8 * lh) * DST + nc] = acc[t][r];
        }
    }
    __syncthreads();

    // ---- Phase 4: tree routing product + GEMM2: out = mu[16x1024] * probs[1024x112]
    v8f acc2 = vzero;
    const int m16 = tid & 15;
    const int jb  = tid >> 4;
    for (int cb = 0; cb < 16; ++cb) {               // 16 chunks of 64 leaves
        // all 256 threads: compute mu chunk into musT[lf][m]
        #pragma unroll
        for (int i = 0; i < 4; ++i) {
            const int lf   = jb + 16 * i;           // 0..63
            const int leaf = cb * 64 + lf;
            float mu = 1.0f;
            #pragma unroll
            for (int l = 0; l < TDEPTH; ++l) {
                const int node = (1 << l) + (leaf >> (TDEPTH - l));
                const float dv = ds[m16 * DST + node];
                mu *= ((leaf >> (TDEPTH - 1 - l)) & 1) ? (1.0f - dv) : dv;
            }
            musT[lf * MUT + m16] = mu;
        }
        __syncthreads();
        if (wave < 7) {                             // wave-uniform: EXEC stays all-1s
            const int n0 = wave * 16 + lm;
            #pragma unroll
            for (int kl = 0; kl < 64; kl += 4) {
                const int ka = kl + 2 * lh;
                v2f a, bf;
                a.x = musT[ka * MUT + lm];
                a.y = musT[(ka + 1) * MUT + lm];
                const int kg = cb * 64 + ka;
                bf.x = probs[kg * NCLSP + n0];
                bf.y = probs[(kg + 1) * NCLSP + n0];
                acc2 = __builtin_amdgcn_wmma_f32_16x16x4_f32(
                    false, a, false, bf, (short)0, acc2, false, false);
            }
        }
        __syncthreads();                            // before next chunk overwrites musT
    }

    // ---- Phase 5: store out[16 x 100]
    if (wave < 7) {
        const int col = wave * 16 + lm;
        if (col < NCLS) {
            #pragma unroll
            for (int r = 0; r < 8; ++r) {
                out[(size_t)(row0 + r + 8 * lh) * NCLS + col] = acc2[r];
            }
        }
    }
}

// ---------------------------------------------------------------------------
extern "C" void kernel_launch(void* const* d_in, const int* in_sizes, int n_in,
                              void* d_out, int out_size, void* d_ws, size_t ws_size,
                              hipStream_t stream) {
    const float* feat = (const float*)d_in[0];   // [16384, 512]
    const float* mask = (const float*)d_in[1];   // [256, 512] one-hot rows
    const float* W    = (const float*)d_in[2];   // [256, 1024]
    const float* b    = (const float*)d_in[3];   // [1024]
    const float* pi   = (const float*)d_in[4];   // [1024, 100]
    float* out = (float*)d_out;                  // [16384, 100]

    int*   idx   = (int*)d_ws;                           // 256 ints
    float* probs = (float*)((char*)d_ws + 1024);         // 1024 x 112 floats

    ndt_extract_idx<<<1, 256, 0, stream>>>(mask, idx);
    ndt_softmax_pi<<<(NLEAF + 255) / 256, 256, 0, stream>>>(pi, probs);
    ndt_main<<<BATCH / 16, 256, 0, stream>>>(feat, W, b, idx, probs, out);
}